// RecursiveLogitRouteChoice_3753801417306
// MI455X (gfx1250) — compile-verified
//
#include <hip/hip_runtime.h>
#include <math.h>

typedef __attribute__((ext_vector_type(2))) float v2f;
typedef __attribute__((ext_vector_type(8))) float v8f;

#define NITERS 100

// ---------------------------------------------------------------------------
// Encoder: rewards = -softplus(feats@W + b), exp_rewards = exp(rewards)
// Uses V_WMMA_F32_16X16X4_F32: 16-edge tile, 8 chained WMMAs cover K=32.
// B holds W broadcast across all 16 columns -> every D column is the result.
// ---------------------------------------------------------------------------
__global__ void rl_encode_wmma(const float* __restrict__ feats,
                               const float* __restrict__ W,
                               const float* __restrict__ bvec,
                               float* __restrict__ rewards,
                               float* __restrict__ expr,
                               int n_edges)
{
    const int lane    = threadIdx.x & 31;
    const int half    = lane >> 4;      // 0: K={0,1}+4s, 1: K={2,3}+4s
    const int m       = lane & 15;      // edge row within tile (A operand)
    const int wid     = (blockIdx.x * blockDim.x + threadIdx.x) >> 5;
    const int wstride = (gridDim.x * blockDim.x) >> 5;
    const int tiles   = n_edges >> 4;
    const float bias  = bvec[0];

    // B operand per WMMA step s: lane needs W[4s + 2*half + {0,1}]
    v2f bw[8];
#pragma unroll
    for (int s = 0; s < 8; ++s) {
        bw[s].x = W[4 * s + 2 * half + 0];
        bw[s].y = W[4 * s + 2 * half + 1];
    }

    for (int t = wid; t < tiles; t += wstride) {
        const float* row = feats + (size_t)(t * 16 + m) * 32;
        v8f c = {};
#pragma unroll
        for (int s = 0; s < 8; ++s) {
            v2f a;
            a.x = row[4 * s + 2 * half + 0];
            a.y = row[4 * s + 2 * half + 1];
            // 8 args: (neg_a, A, neg_b, B, c_mod, C, reuse_a, reuse_b)
            c = __builtin_amdgcn_wmma_f32_16x16x4_f32(
                    false, a, false, bw[s], (short)0, c, false, false);
        }
        // D[r][n] identical over n. lane0: M=0..7 in c[0..7]; lane16: M=8..15.
        const int base = t * 16;
        if (lane == 0) {
#pragma unroll
            for (int r = 0; r < 8; ++r) {
                float y  = c[r] + bias;
                float sp = fmaxf(y, 0.0f) + log1pf(expf(-fabsf(y)));
                rewards[base + r] = -sp;
                expr[base + r]    = expf(-sp);
            }
        } else if (lane == 16) {
#pragma unroll
            for (int r = 0; r < 8; ++r) {
                float y  = c[r] + bias;
                float sp = fmaxf(y, 0.0f) + log1pf(expf(-fabsf(y)));
                rewards[base + 8 + r] = -sp;
                expr[base + 8 + r]    = expf(-sp);
            }
        }
    }

    // scalar tail for n_edges % 16 (E=1.6M -> empty; kept for generality)
    const int tail = tiles << 4;
    const int gtid = blockIdx.x * blockDim.x + threadIdx.x;
    for (int e = tail + gtid; e < n_edges; e += gridDim.x * blockDim.x) {
        const float* r2 = feats + (size_t)e * 32;
        float y = bias;
        for (int k = 0; k < 32; ++k) y += r2[k] * W[k];
        float sp = fmaxf(y, 0.0f) + log1pf(expf(-fabsf(y)));
        rewards[e] = -sp;
        expr[e]    = expf(-sp);
    }
}

// ---------------------------------------------------------------------------
// CSR build: histogram -> single-block scan -> fill (payload = {dst, exp_r})
// ---------------------------------------------------------------------------
__global__ void rl_zero_u32(unsigned* __restrict__ p, int n)
{
    int i = blockIdx.x * blockDim.x + threadIdx.x;
    if (i < n) p[i] = 0u;
}

__global__ void rl_hist(const int* __restrict__ src,
                        unsigned* __restrict__ counts, int E)
{
    int e = blockIdx.x * blockDim.x + threadIdx.x;
    if (e < E) atomicAdd(&counts[src[e]], 1u);
}

__global__ void rl_scan_rowptr(const unsigned* __restrict__ counts,
                               unsigned* __restrict__ row_ptr,
                               unsigned* __restrict__ cursor, int n)
{
    __shared__ unsigned s[1024];
    const int t     = threadIdx.x;
    const int chunk = (n + 1023) / 1024;
    const int lo    = t * chunk;
    const int hi    = (lo + chunk < n) ? (lo + chunk) : n;

    unsigned part = 0;
    for (int i = lo; i < hi; ++i) part += counts[i];
    s[t] = part;
    __syncthreads();
    for (int off = 1; off < 1024; off <<= 1) {   // inclusive Hillis-Steele
        unsigned v = (t >= off) ? s[t - off] : 0u;
        __syncthreads();
        s[t] += v;
        __syncthreads();
    }
    unsigned run = s[t] - part;                  // exclusive prefix
    for (int i = lo; i < hi; ++i) {
        row_ptr[i] = run;
        cursor[i]  = run;
        run += counts[i];
    }
    if (t == 1023) row_ptr[n] = s[1023];
}

__global__ void rl_csr_fill(const int* __restrict__ src,
                            const int* __restrict__ dst,
                            const float* __restrict__ expr,
                            unsigned* __restrict__ cursor,
                            int2* __restrict__ payload, int E)
{
    int e = blockIdx.x * blockDim.x + threadIdx.x;
    if (e >= E) return;
    unsigned pos = atomicAdd(&cursor[src[e]], 1u);
    payload[pos] = make_int2(dst[e], __float_as_int(expr[e]));
}

// ---------------------------------------------------------------------------
// Fixed-point iteration (L2-resident gather, no atomics). Clamp via mask.
// ---------------------------------------------------------------------------
__global__ void rl_init_x(const unsigned char* __restrict__ mask,
                          float* __restrict__ x, int n)
{
    int i = blockIdx.x * blockDim.x + threadIdx.x;
    if (i < n) x[i] = mask[i] ? 1.0f : 0.0f;
}

__global__ void rl_vi_step(const unsigned* __restrict__ row_ptr,
                           const int2* __restrict__ payload,
                           const unsigned char* __restrict__ mask,
                           const float* __restrict__ xin,
                           float* __restrict__ xout, int n)
{
    int i = blockIdx.x * blockDim.x + threadIdx.x;
    if (i >= n) return;
    if (mask[i]) { xout[i] = 1.0f; return; }
    const unsigned lo = row_ptr[i], hi = row_ptr[i + 1];
    float sum = 0.0f;
    for (unsigned j = lo; j < hi; ++j) {
        int2 p = payload[j];                      // single b64 load
        sum += __int_as_float(p.y) * xin[p.x];
    }
    xout[i] = sum;
}

__global__ void rl_values(const float* __restrict__ x,
                          float* __restrict__ values, int n)
{
    int i = blockIdx.x * blockDim.x + threadIdx.x;
    if (i < n) values[i] = logf(x[i]);
}

__global__ void rl_edge_probs(const int* __restrict__ src,
                              const int* __restrict__ dst,
                              const float* __restrict__ expr,
                              const float* __restrict__ x,
                              float* __restrict__ probs, int E)
{
    int e = blockIdx.x * blockDim.x + threadIdx.x;
    if (e < E) probs[e] = expr[e] * x[dst[e]] / x[src[e]];
}

// ---------------------------------------------------------------------------
extern "C" void kernel_launch(void* const* d_in, const int* in_sizes, int n_in,
                              void* d_out, int out_size, void* d_ws, size_t ws_size,
                              hipStream_t stream)
{
    const int E = in_sizes[0] / 2;   // edge_index [2,E]
    const int N = in_sizes[2];       // sink_node_mask [N]

    const int*           src   = (const int*)d_in[0];
    const int*           dst   = (const int*)d_in[0] + E;
    const float*         feats = (const float*)d_in[1];
    const unsigned char* mask  = (const unsigned char*)d_in[2];
    const float*         W     = (const float*)d_in[3];
    const float*         bvec  = (const float*)d_in[4];

    float* out_rewards = (float*)d_out;              // [E]
    float* out_values  = (float*)d_out + E;          // [N]
    float* out_probs   = (float*)d_out + E + N;      // [E]

    // Workspace carve-up (256B aligned)
    char*  w = (char*)d_ws;
    size_t o = 0;
    auto carve = [&](size_t bytes) { void* p = w + o; o = (o + bytes + 255) & ~(size_t)255; return p; };
    float*    expr    = (float*)   carve((size_t)E * 4);
    float*    x0      = (float*)   carve((size_t)N * 4);
    float*    x1      = (float*)   carve((size_t)N * 4);
    unsigned* counts  = (unsigned*)carve((size_t)N * 4);
    unsigned* row_ptr = (unsigned*)carve((size_t)(N + 1) * 4);
    unsigned* cursor  = (unsigned*)carve((size_t)N * 4);
    int2*     payload = (int2*)    carve((size_t)E * 8);
    (void)ws_size; (void)n_in; (void)out_size;

    const int TPB   = 256;
    const int gE    = (E + TPB - 1) / TPB;
    const int gN    = (N + TPB - 1) / TPB;
    const int gEnc  = 2048;   // grid-stride over E/16 tiles, 8 waves/block

    // 1) Encoder (WMMA f32 16x16x4)
    rl_encode_wmma<<<gEnc, TPB, 0, stream>>>(feats, W, bvec, out_rewards, expr, E);

    // 2) CSR build
    rl_zero_u32<<<gN, TPB, 0, stream>>>(counts, N);
    rl_hist<<<gE, TPB, 0, stream>>>(src, counts, E);
    rl_scan_rowptr<<<1, 1024, 0, stream>>>(counts, row_ptr, cursor, N);
    rl_csr_fill<<<gE, TPB, 0, stream>>>(src, dst, expr, cursor, payload, E);

    // 3) 100 value-iteration steps (ping-pong, all state L2-resident)
    rl_init_x<<<gN, TPB, 0, stream>>>(mask, x0, N);
    float* xin = x0; float* xout = x1;
    for (int it = 0; it < NITERS; ++it) {
        rl_vi_step<<<gN, TPB, 0, stream>>>(row_ptr, payload, mask, xin, xout, N);
        float* tmp = xin; xin = xout; xout = tmp;
    }
    // final exp_values now in xin

    // 4) Epilogue
    rl_values<<<gN, TPB, 0, stream>>>(xin, out_values, N);
    rl_edge_probs<<<gE, TPB, 0, stream>>>(src, dst, expr, xin, out_probs, E);
}